// BasicBlock_53730040873330
// MI455X (gfx1250) — compile-verified
//
#include <hip/hip_runtime.h>
#include <math.h>

// Problem constants (from reference)
#define L_SEQ   16384      // 16*32*32
#define CCH     32         // per-branch channels
#define NST     64         // d_state
#define NCHUNK  128        // scan chunks
#define LCHUNK  128        // steps per chunk (NCHUNK*LCHUNK == L_SEQ)

typedef float v2f __attribute__((ext_vector_type(2)));
typedef float v8f __attribute__((ext_vector_type(8)));

__device__ __forceinline__ float sigmoidf_(float x) { return 1.0f / (1.0f + __expf(-x)); }
__device__ __forceinline__ float siluf_(float x)    { return x * sigmoidf_(x); }
__device__ __forceinline__ float geluf_(float x) {
    // jax.nn.gelu (approximate=True, tanh form)
    float x3 = x * x * x;
    return 0.5f * x * (1.0f + tanhf(0.7978845608028654f * (x + 0.044715f * x3)));
}
__device__ __forceinline__ float softplusf_(float x) {
    return (x > 20.0f) ? x : log1pf(__expf(x));
}
__device__ __forceinline__ float wave_sum32(float v) {
    #pragma unroll
    for (int off = 16; off > 0; off >>= 1) v += __shfl_xor(v, off, 32);
    return v;
}

// ---------------------------------------------------------------------------
// f32 WMMA GEMM:  Out[M,N] = act(A[M,K] @ W[N,K]^T + bias) (+ Out residual)
// One wave per 16x16 output tile using V_WMMA_F32_16X16X4_F32.
// A layout (16x4, MxK): lanes 0-15 -> M=lane, K=k0..k0+1; lanes 16-31 -> K=k0+2..k0+3
// B layout (4x16, KxN): lane n holds B[k][n] = W[n][k] pairs, same K split.
// Out-of-range N lanes read a clamped (in-bounds) weight row; their columns
// are garbage but never stored, so no predication is needed in the K loop.
// ---------------------------------------------------------------------------
template<int ACT, bool ADDRES>
__global__ void __launch_bounds__(32) gemm_wmma_f32(
    const float* __restrict__ A, const float* __restrict__ W,
    const float* __restrict__ bias, float* __restrict__ Out,
    int M, int N, int K, int lda, int ldw, int ldo)
{
    int m0   = blockIdx.x * 16;
    int n0   = blockIdx.y * 16;
    int lane = threadIdx.x;        // 0..31
    int half = lane >> 4;          // 0 or 1 (K sub-pair)
    int mr   = lane & 15;

    v8f acc = {};
    const float* Arow = A + (size_t)(m0 + mr) * lda + 2 * half;
    int  n  = n0 + mr;
    bool nv = (n < N);
    const float* Wrow = W + (size_t)(nv ? n : (N - 1)) * ldw + 2 * half;

    #pragma unroll 4
    for (int k0 = 0; k0 < K; k0 += 4) {
        v2f a = *(const v2f*)(Arow + k0);
        v2f b = *(const v2f*)(Wrow + k0);
        acc = __builtin_amdgcn_wmma_f32_16x16x4_f32(
                  false, a, false, b, (short)0, acc, false, false);
    }

    if (nv) {
        float bv = bias ? bias[n] : 0.0f;
        float* Ocol = Out + (size_t)(m0 + 8 * half) * ldo + n;
        #pragma unroll
        for (int v = 0; v < 8; ++v) {
            float val = acc[v] + bv;
            if (ACT == 1) val = geluf_(val);
            size_t idx = (size_t)v * ldo;
            if (ADDRES) val += Ocol[idx];
            Ocol[idx] = val;
        }
    }
}

// ---------------------------------------------------------------------------
// LayerNorm over C=32 per token; lane = channel, wave per token (coalesced rows)
// ---------------------------------------------------------------------------
__global__ void ln_kernel(const float* __restrict__ x, const float* __restrict__ w,
                          const float* __restrict__ b, float* __restrict__ out, int Ltok)
{
    int wave   = (blockIdx.x * blockDim.x + threadIdx.x) >> 5;
    int lane   = threadIdx.x & 31;
    int nwaves = (gridDim.x * blockDim.x) >> 5;
    float wc = w[lane], bc = b[lane];
    for (int t = wave; t < Ltok; t += nwaves) {
        float v   = x[(size_t)t * 32 + lane];
        float m   = wave_sum32(v) * (1.0f / 32.0f);
        float d   = v - m;
        float var = wave_sum32(d * d) * (1.0f / 32.0f);
        out[(size_t)t * 32 + lane] = d * rsqrtf(var + 1e-5f) * wc + bc;
    }
}

// ---------------------------------------------------------------------------
// Depthwise causal conv1d (kernel 4) + SiLU. Input is xc half of xz [L,64].
// ---------------------------------------------------------------------------
__global__ void dwconv_silu_kernel(const float* __restrict__ xz, const float* __restrict__ w,
                                   const float* __restrict__ b, float* __restrict__ out)
{
    int t = blockIdx.x * blockDim.x + threadIdx.x;     // l*32 + c
    if (t >= L_SEQ * CCH) return;
    int c = t & 31;
    int l = t >> 5;
    float acc = b[c];
    #pragma unroll
    for (int j = 0; j < 4; ++j) {
        int ls = l - 3 + j;
        if (ls >= 0) acc += w[c * 4 + j] * xz[(size_t)ls * 64 + c];
    }
    out[t] = siluf_(acc);
}

// ---------------------------------------------------------------------------
// delta = softplus(dt @ dt_w^T + dt_b), dt = dbl[:, 0:2]
// ---------------------------------------------------------------------------
__global__ void delta_kernel(const float* __restrict__ dbl, const float* __restrict__ dtw,
                             const float* __restrict__ dtb, float* __restrict__ out)
{
    int t = blockIdx.x * blockDim.x + threadIdx.x;
    if (t >= L_SEQ * CCH) return;
    int c = t & 31;
    int l = t >> 5;
    float d0 = dbl[(size_t)l * 130 + 0], d1 = dbl[(size_t)l * 130 + 1];
    out[t] = softplusf_(d0 * dtw[c * 2 + 0] + d1 * dtw[c * 2 + 1] + dtb[c]);
}

// ---------------------------------------------------------------------------
// Selective scan, 3-phase chunked parallel scan.
// Phase 1: per-(chunk,c) wave (lane holds states n, n+32): local scan from 0,
//          record P = prod(a) and local end-state h.
// ---------------------------------------------------------------------------
__global__ void scan_phase1(const float* __restrict__ delta, const float* __restrict__ xc,
                            const float* __restrict__ dbl, const float* __restrict__ A_log,
                            float* __restrict__ Pbuf, float* __restrict__ Hbuf)
{
    int chunk = blockIdx.x;
    int c     = blockIdx.y;
    int lane  = threadIdx.x;
    int n1 = lane, n2 = lane + 32;
    float An1 = -__expf(A_log[c * 64 + n1]);
    float An2 = -__expf(A_log[c * 64 + n2]);
    float h1 = 0.f, h2 = 0.f, P1 = 1.f, P2 = 1.f;
    int l0 = chunk * LCHUNK;
    for (int l = l0; l < l0 + LCHUNK; ++l) {
        float d  = delta[(size_t)l * 32 + c];
        float x  = xc[(size_t)l * 32 + c];
        float dx = d * x;
        const float* B = dbl + (size_t)l * 130 + 2;
        float a1 = __expf(d * An1), a2 = __expf(d * An2);
        h1 = a1 * h1 + dx * B[n1];
        h2 = a2 * h2 + dx * B[n2];
        P1 *= a1; P2 *= a2;
    }
    size_t base = (size_t)chunk * 2048 + c * 64;
    Pbuf[base + n1] = P1; Pbuf[base + n2] = P2;
    Hbuf[base + n1] = h1; Hbuf[base + n2] = h2;
}

// Phase 2: serial scan across chunks; 2048 independent (c,n) threads.
__global__ void scan_phase2(const float* __restrict__ Pbuf, const float* __restrict__ Hbuf,
                            float* __restrict__ Hinit)
{
    int t = blockIdx.x * blockDim.x + threadIdx.x;
    if (t >= 2048) return;
    float carry = 0.0f;
    for (int j = 0; j < NCHUNK; ++j) {
        size_t idx = (size_t)j * 2048 + t;
        Hinit[idx] = carry;
        carry = Pbuf[idx] * carry + Hbuf[idx];
    }
}

// Phase 3: replay each chunk with corrected initial state; fuse D-skip and
// silu(z) gate into the epilogue: u = (y + xc*D) * silu(z)
__global__ void scan_phase3(const float* __restrict__ delta, const float* __restrict__ xc,
                            const float* __restrict__ dbl, const float* __restrict__ A_log,
                            const float* __restrict__ Hinit, const float* __restrict__ Dskip,
                            const float* __restrict__ xz, float* __restrict__ u)
{
    int chunk = blockIdx.x;
    int c     = blockIdx.y;
    int lane  = threadIdx.x;
    int n1 = lane, n2 = lane + 32;
    float An1 = -__expf(A_log[c * 64 + n1]);
    float An2 = -__expf(A_log[c * 64 + n2]);
    size_t base = (size_t)chunk * 2048 + c * 64;
    float h1 = Hinit[base + n1], h2 = Hinit[base + n2];
    float Dc = Dskip[c];
    int l0 = chunk * LCHUNK;
    for (int l = l0; l < l0 + LCHUNK; ++l) {
        float d  = delta[(size_t)l * 32 + c];
        float x  = xc[(size_t)l * 32 + c];
        float dx = d * x;
        const float* B  = dbl + (size_t)l * 130 + 2;
        const float* Cm = dbl + (size_t)l * 130 + 66;
        float a1 = __expf(d * An1), a2 = __expf(d * An2);
        h1 = a1 * h1 + dx * B[n1];
        h2 = a2 * h2 + dx * B[n2];
        float y = wave_sum32(Cm[n1] * h1 + Cm[n2] * h2);
        if (lane == 0) {
            y += x * Dc;
            float z = xz[(size_t)l * 64 + 32 + c];
            u[(size_t)l * 32 + c] = y * siluf_(z);
        }
    }
}

// ---------------------------------------------------------------------------
// Fused conv3d(3x3x3, pad 1) + InstanceNorm + LeakyReLU (+ optional skip add)
// Block per output channel; two-pass with LDS mean/var reduction.
// ---------------------------------------------------------------------------
template<bool ADD_SKIP>
__global__ void conv_in_lrelu(const float* __restrict__ in, const float* __restrict__ w,
                              const float* __restrict__ bias, const float* __restrict__ skip,
                              float* __restrict__ out)
{
    const int D = 16, H = 32, Wd = 32, SP = 16384;
    int oc  = blockIdx.x;
    int tid = threadIdx.x;
    float s = 0.f, s2 = 0.f;
    for (int sp = tid; sp < SP; sp += 256) {
        int d = sp >> 10, h = (sp >> 5) & 31, xw = sp & 31;
        float acc = bias[oc];
        for (int ic = 0; ic < 32; ++ic) {
            const float* wp = w + (size_t)(oc * 32 + ic) * 27;
            const float* ip = in + (size_t)ic * SP;
            #pragma unroll
            for (int kd = 0; kd < 3; ++kd) {
                int dd = d + kd - 1;
                if (dd < 0 || dd >= D) continue;
                #pragma unroll
                for (int kh = 0; kh < 3; ++kh) {
                    int hh = h + kh - 1;
                    if (hh < 0 || hh >= H) continue;
                    #pragma unroll
                    for (int kw = 0; kw < 3; ++kw) {
                        int ww = xw + kw - 1;
                        if (ww < 0 || ww >= Wd) continue;
                        acc += wp[kd * 9 + kh * 3 + kw] * ip[(dd << 10) + (hh << 5) + ww];
                    }
                }
            }
        }
        out[(size_t)oc * SP + sp] = acc;
        s += acc; s2 += acc * acc;
    }
    __shared__ float sh1[256], sh2[256];
    sh1[tid] = s; sh2[tid] = s2;
    __syncthreads();
    for (int off = 128; off > 0; off >>= 1) {
        if (tid < off) { sh1[tid] += sh1[tid + off]; sh2[tid] += sh2[tid + off]; }
        __syncthreads();
    }
    float mean = sh1[0] * (1.0f / 16384.0f);
    float var  = sh2[0] * (1.0f / 16384.0f) - mean * mean;
    float inv  = rsqrtf(var + 1e-5f);
    for (int sp = tid; sp < SP; sp += 256) {
        size_t idx = (size_t)oc * SP + sp;
        float v = (out[idx] - mean) * inv;
        v = (v >= 0.0f) ? v : 0.01f * v;
        if (ADD_SKIP) v += skip[idx];
        out[idx] = v;
    }
}

// ---------------------------------------------------------------------------
// Tiled transpose: in[R,C] -> out[C,R]
// ---------------------------------------------------------------------------
__global__ void transpose_kernel(const float* __restrict__ in, float* __restrict__ out,
                                 int R, int C)
{
    __shared__ float tile[32][33];
    int bx = blockIdx.x * 32;   // col base
    int by = blockIdx.y * 32;   // row base
    int tx = threadIdx.x, ty = threadIdx.y;
    for (int j = ty; j < 32; j += 8) {
        int r = by + j, c = bx + tx;
        if (r < R && c < C) tile[j][tx] = in[(size_t)r * C + c];
    }
    __syncthreads();
    for (int j = ty; j < 32; j += 8) {
        int c = bx + j, r = by + tx;
        if (r < R && c < C) out[(size_t)c * R + r] = tile[tx][j];
    }
}

// ---------------------------------------------------------------------------
// MAFM gated fusion: gate = sigmoid(1x1x1 conv over cat), out = gate*cat + res
// Thread per voxel; 64-channel dot fully register-resident.
// ---------------------------------------------------------------------------
__global__ void fuse_kernel(const float* __restrict__ x1, const float* __restrict__ x2,
                            const float* __restrict__ fw, const float* __restrict__ fb,
                            const float* __restrict__ resid, float* __restrict__ out)
{
    int l = blockIdx.x * blockDim.x + threadIdx.x;
    if (l >= L_SEQ) return;
    float cat[64];
    #pragma unroll
    for (int i = 0; i < 32; ++i) cat[i] = x1[(size_t)i * L_SEQ + l];
    #pragma unroll
    for (int i = 0; i < 32; ++i) cat[32 + i] = x2[(size_t)i * L_SEQ + l];
    #pragma unroll
    for (int o = 0; o < 64; ++o) {
        float acc = fb[o];
        #pragma unroll
        for (int i = 0; i < 64; ++i) acc += fw[o * 64 + i] * cat[i];
        float g = 1.0f / (1.0f + __expf(-acc));
        out[(size_t)o * L_SEQ + l] = g * cat[o] + resid[(size_t)o * L_SEQ + l];
    }
}

// ---------------------------------------------------------------------------
extern "C" void kernel_launch(void* const* d_in, const int* in_sizes, int n_in,
                              void* d_out, int out_size, void* d_ws, size_t ws_size,
                              hipStream_t stream)
{
    (void)in_sizes; (void)n_in; (void)out_size; (void)ws_size;
    const float* x        = (const float*)d_in[0];
    const float* conv1_w  = (const float*)d_in[1];
    const float* conv1_b  = (const float*)d_in[2];
    const float* conv2_w  = (const float*)d_in[3];
    const float* conv2_b  = (const float*)d_in[4];
    const float* ln1_w    = (const float*)d_in[5];
    const float* ln1_b    = (const float*)d_in[6];
    const float* in_proj  = (const float*)d_in[7];
    const float* c1d_w    = (const float*)d_in[8];
    const float* c1d_b    = (const float*)d_in[9];
    const float* xp_w     = (const float*)d_in[10];
    const float* dt_w     = (const float*)d_in[11];
    const float* dt_b     = (const float*)d_in[12];
    const float* A_log    = (const float*)d_in[13];
    const float* D_skip   = (const float*)d_in[14];
    const float* out_w    = (const float*)d_in[15];
    const float* ln2_w    = (const float*)d_in[16];
    const float* ln2_b    = (const float*)d_in[17];
    const float* mlp_w1   = (const float*)d_in[18];
    const float* mlp_b1   = (const float*)d_in[19];
    const float* mlp_w2   = (const float*)d_in[20];
    const float* mlp_b2   = (const float*)d_in[21];
    const float* fuse_w   = (const float*)d_in[22];
    const float* fuse_b   = (const float*)d_in[23];
    float* outp = (float*)d_out;

    const size_t L = L_SEQ;
    float* ws      = (float*)d_ws;            // ~41 MB total
    float* ws_h    = ws;                      // [32,L]  conv1 out
    float* ws_x1   = ws_h    + 32 * L;        // [32,L]  conv branch out
    float* ws_seq  = ws_x1   + 32 * L;        // [L,32]  mamba residual stream
    float* ws_ln   = ws_seq  + 32 * L;        // [L,32]
    float* ws_xz   = ws_ln   + 32 * L;        // [L,64]
    float* ws_xc   = ws_xz   + 64 * L;        // [L,32]  conv1d+silu out
    float* ws_dbl  = ws_xc   + 32 * L;        // [L,130] dt|B|C
    float* ws_del  = ws_dbl  + 130 * L;       // [L,32]  delta
    float* ws_u    = ws_del  + 32 * L;        // [L,32]  gated scan out
    float* ws_mlp  = ws_u    + 32 * L;        // [L,128]
    float* ws_P    = ws_mlp  + 128 * L;       // [NCHUNK,2048]
    float* ws_hf   = ws_P    + (size_t)NCHUNK * 2048;
    float* ws_hi   = ws_hf   + (size_t)NCHUNK * 2048;
    float* ws_x2   = ws_hi   + (size_t)NCHUNK * 2048;  // [32,L]

    // ---- dense conv branch: h = lrelu(IN(conv1(x1))); x1' = h + lrelu(IN(conv2(h)))
    conv_in_lrelu<false><<<dim3(32), dim3(256), 0, stream>>>(x,     conv1_w, conv1_b, nullptr, ws_h);
    conv_in_lrelu<true ><<<dim3(32), dim3(256), 0, stream>>>(ws_h,  conv2_w, conv2_b, ws_h,    ws_x1);

    // ---- mamba branch init: seq[l,c] = x2[c,l]
    transpose_kernel<<<dim3((L_SEQ + 31) / 32, 1), dim3(32, 8), 0, stream>>>(x + 32 * L, ws_seq, 32, L_SEQ);

    for (int i = 0; i < 4; ++i) {
        const float* l1w = ln1_w + i * 32;    const float* l1b = ln1_b + i * 32;
        const float* ipw = in_proj + (size_t)i * 64 * 32;
        const float* cw  = c1d_w + (size_t)i * 32 * 4;   const float* cb = c1d_b + i * 32;
        const float* xpw = xp_w  + (size_t)i * 130 * 32;
        const float* dtw = dt_w  + (size_t)i * 32 * 2;   const float* dtb = dt_b + i * 32;
        const float* Al  = A_log + (size_t)i * 32 * 64;  const float* Ds  = D_skip + i * 32;
        const float* ow  = out_w + (size_t)i * 32 * 32;
        const float* l2w = ln2_w + i * 32;    const float* l2b = ln2_b + i * 32;
        const float* w1  = mlp_w1 + (size_t)i * 128 * 32; const float* b1 = mlp_b1 + i * 128;
        const float* w2  = mlp_w2 + (size_t)i * 32 * 128; const float* b2 = mlp_b2 + i * 32;

        ln_kernel<<<256, 256, 0, stream>>>(ws_seq, l1w, l1b, ws_ln, L_SEQ);
        gemm_wmma_f32<0, false><<<dim3(L_SEQ / 16, 4), 32, 0, stream>>>(
            ws_ln, ipw, nullptr, ws_xz, L_SEQ, 64, 32, 32, 32, 64);
        dwconv_silu_kernel<<<(L_SEQ * 32 + 255) / 256, 256, 0, stream>>>(ws_xz, cw, cb, ws_xc);
        gemm_wmma_f32<0, false><<<dim3(L_SEQ / 16, 9), 32, 0, stream>>>(
            ws_xc, xpw, nullptr, ws_dbl, L_SEQ, 130, 32, 32, 32, 130);
        delta_kernel<<<(L_SEQ * 32 + 255) / 256, 256, 0, stream>>>(ws_dbl, dtw, dtb, ws_del);
        scan_phase1<<<dim3(NCHUNK, 32), 32, 0, stream>>>(ws_del, ws_xc, ws_dbl, Al, ws_P, ws_hf);
        scan_phase2<<<8, 256, 0, stream>>>(ws_P, ws_hf, ws_hi);
        scan_phase3<<<dim3(NCHUNK, 32), 32, 0, stream>>>(ws_del, ws_xc, ws_dbl, Al, ws_hi, Ds, ws_xz, ws_u);
        // seq += u @ out_w^T
        gemm_wmma_f32<0, true><<<dim3(L_SEQ / 16, 2), 32, 0, stream>>>(
            ws_u, ow, nullptr, ws_seq, L_SEQ, 32, 32, 32, 32, 32);
        // MLP block: seq += gelu(ln2(seq) @ w1^T + b1) @ w2^T + b2
        ln_kernel<<<256, 256, 0, stream>>>(ws_seq, l2w, l2b, ws_ln, L_SEQ);
        gemm_wmma_f32<1, false><<<dim3(L_SEQ / 16, 8), 32, 0, stream>>>(
            ws_ln, w1, b1, ws_mlp, L_SEQ, 128, 32, 32, 32, 128);
        gemm_wmma_f32<0, true><<<dim3(L_SEQ / 16, 2), 32, 0, stream>>>(
            ws_mlp, w2, b2, ws_seq, L_SEQ, 32, 128, 128, 128, 32);
    }

    // ---- back to channel-major, then gated fusion + outer residual
    transpose_kernel<<<dim3(1, (L_SEQ + 31) / 32), dim3(32, 8), 0, stream>>>(ws_seq, ws_x2, L_SEQ, 32);
    fuse_kernel<<<(L_SEQ + 127) / 128, 128, 0, stream>>>(ws_x1, ws_x2, fuse_w, fuse_b, x, outp);
}